// MahalanobisDistanceLayer_86861418594470
// MI455X (gfx1250) — compile-verified
//
#include <hip/hip_runtime.h>
#include <math.h>

typedef __attribute__((ext_vector_type(2))) float v2f;
typedef __attribute__((ext_vector_type(8))) float v8f;

#define B_ROWS 8192
#define D_FEAT 64
#define NELEM (B_ROWS * D_FEAT)      // 524288 per matrix
#define EPS_IN 1e-4f
#define EPS_OUT 1e-6f
#define COL_CHUNKS 8
#define TILES_PER_CHUNK (512 / COL_CHUNKS)  // 64 column tiles per chunk
#define LDS_STRIDE 68                       // 64 + 4 pad -> conflict-free v2f reads

// ---------------------------------------------------------------------------
// Kernel 1: per-feature variance of concat(a, b) + EPS_IN over 2B samples.
// Double accumulation avoids E[x^2]-E[x]^2 cancellation. Stores 1/var.
// grid = (64 features, 2 pairs), block = 256
// ---------------------------------------------------------------------------
__global__ void var_kernel(const float* __restrict__ a,
                           const float* __restrict__ pos,
                           const float* __restrict__ neg,
                           float* __restrict__ rvar) {
  const int f = blockIdx.x;
  const int pair = blockIdx.y;
  const float* b = pair ? neg : pos;
  const int t = threadIdx.x;
  double s = 0.0, s2 = 0.0;
  for (int r = t; r < B_ROWS; r += 256) {
    float xa = a[r * D_FEAT + f] + EPS_IN;
    float xb = b[r * D_FEAT + f] + EPS_IN;
    s  += (double)xa + (double)xb;
    s2 += (double)xa * (double)xa + (double)xb * (double)xb;
  }
  __shared__ double ls[256];
  __shared__ double ls2[256];
  ls[t] = s; ls2[t] = s2;
  __syncthreads();
  for (int o = 128; o > 0; o >>= 1) {
    if (t < o) { ls[t] += ls[t + o]; ls2[t] += ls2[t + o]; }
    __syncthreads();
  }
  if (t == 0) {
    const double n = 2.0 * (double)B_ROWS;
    double mean = ls[0] / n;
    double var = ls2[0] / n - mean * mean;
    rvar[pair * D_FEAT + f] = (float)(1.0 / var);
  }
}

// ---------------------------------------------------------------------------
// Kernel 2: diff = a - b ; sdiff = diff * (1/var), both pairs.
// ---------------------------------------------------------------------------
__global__ void diff_kernel(const float* __restrict__ a,
                            const float* __restrict__ p,
                            const float* __restrict__ n,
                            const float* __restrict__ rvar,
                            float* __restrict__ sdiff,
                            float* __restrict__ diffb) {
  const int idx = blockIdx.x * 256 + threadIdx.x;
  const int f = idx & (D_FEAT - 1);
  const float av = a[idx];
  const float d1 = av - p[idx];
  const float d2 = av - n[idx];
  diffb[idx] = d1;
  diffb[NELEM + idx] = d2;
  sdiff[idx] = d1 * rvar[f];
  sdiff[NELEM + idx] = d2 * rvar[D_FEAT + f];
}

// ---------------------------------------------------------------------------
// Kernel 3: fused GEMM (sdiff @ diff^T) + sqrt + row-sum via fp32 WMMA.
// block = 256 threads = 8 wave32s, covers 128 rows; iterates 64 column tiles.
// B tiles double-buffered through padded LDS: one barrier per tile, global
// prefetch of tile i+1 overlapped with the 16 WMMAs of tile i.
// grid = (64 row-blocks, 8 col-chunks, 2 pairs)
// ---------------------------------------------------------------------------
__global__ void __launch_bounds__(256) mdl_wmma_kernel(
    const float* __restrict__ sdiff, const float* __restrict__ diffb,
    float* __restrict__ partial) {
  const int pair = blockIdx.z;
  const int chunk = blockIdx.y;
  const int rowblk = blockIdx.x;  // 128 rows each
  const float* A  = sdiff + (size_t)pair * NELEM;
  const float* Bm = diffb + (size_t)pair * NELEM;

  const int t = threadIdx.x;
  const int wave = t >> 5;
  const int lane = t & 31;
  const int half = lane >> 4;     // selects K pair within 4-wide K block
  const int m = lane & 15;        // M index for A operand, N index for B operand

  // A operand: this wave's 16 rows, kept in registers for all column tiles.
  // ISA layout (32-bit A 16x4): lanes 0-15 hold {K=0,K=1}, lanes 16-31 {K=2,K=3}.
  const int arow = rowblk * 128 + wave * 16 + m;
  const float* aptr = A + (size_t)arow * D_FEAT;
  v2f areg[16];
#pragma unroll
  for (int s = 0; s < 16; ++s)
    areg[s] = *(const v2f*)(aptr + 4 * s + 2 * half);

  __shared__ float bt[2][16 * LDS_STRIDE];

  const int br = t >> 4;          // tile row this thread stages
  const int bk = (t & 15) * 4;    // feature offset (float4 per thread)
  const int ct_begin = chunk * TILES_PER_CHUNK;

  // Preload first B tile (16x64 floats, coalesced float4 per thread).
  {
    const float4 bv =
        *(const float4*)(Bm + (size_t)ct_begin * 16 * D_FEAT + 4 * t);
    bt[0][br * LDS_STRIDE + bk + 0] = bv.x;
    bt[0][br * LDS_STRIDE + bk + 1] = bv.y;
    bt[0][br * LDS_STRIDE + bk + 2] = bv.z;
    bt[0][br * LDS_STRIDE + bk + 3] = bv.w;
  }
  __syncthreads();

  float rs[8];
#pragma unroll
  for (int v = 0; v < 8; ++v) rs[v] = 0.0f;

  int cur = 0;
  for (int i = 0; i < TILES_PER_CHUNK; ++i) {
    // Prefetch next tile from global while this tile's WMMAs run (uniform cond).
    float4 nv;
    const bool have_next = (i + 1 < TILES_PER_CHUNK);
    if (have_next)
      nv = *(const float4*)(Bm +
                            (size_t)(ct_begin + i + 1) * 16 * D_FEAT + 4 * t);

    const float* btc = bt[cur];
    v8f c = {};
#pragma unroll
    for (int s = 0; s < 16; ++s) {
      // B operand (4x16): lanes 0-15 hold {K=0,K=1} of column n=m, etc.
      // B[k][n] = diff[n][k] -> contiguous v2f from LDS row m.
      v2f breg = *(const v2f*)(&btc[m * LDS_STRIDE + 4 * s + 2 * half]);
      c = __builtin_amdgcn_wmma_f32_16x16x4_f32(
          /*neg_a=*/false, areg[s], /*neg_b=*/false, breg,
          /*c_mod=*/(short)0, c, /*reuse_a=*/false, /*reuse_b=*/false);
    }
    // C/D layout: lane holds N=m; VGPR v holds M = v + 8*half.
    // sqrt(max(q,0)) == reference's where(q<0, 0, sqrt(q)); single
    // v_max_f32 + v_sqrt_f32, no EXEC-modifying branches near the WMMAs.
#pragma unroll
    for (int v = 0; v < 8; ++v)
      rs[v] += __builtin_amdgcn_sqrtf(fmaxf(c[v], 0.0f)) + EPS_OUT;

    if (have_next) {
      float* dst = &bt[cur ^ 1][br * LDS_STRIDE + bk];
      dst[0] = nv.x; dst[1] = nv.y; dst[2] = nv.z; dst[3] = nv.w;
    }
    __syncthreads();   // one barrier per tile (covers buffer handoff)
    cur ^= 1;
  }

  // Reduce over the 16 N-lanes of each half-wave (masks 1,2,4,8 stay in-half).
#pragma unroll
  for (int v = 0; v < 8; ++v) {
    float x = rs[v];
    x += __shfl_xor(x, 1, 32);
    x += __shfl_xor(x, 2, 32);
    x += __shfl_xor(x, 4, 32);
    x += __shfl_xor(x, 8, 32);
    rs[v] = x;
  }
  if (m == 0) {
    // lane 0 -> rows base+0..7, lane 16 -> rows base+8..15
    const int rowbase = rowblk * 128 + wave * 16 + half * 8;
    float* pp = partial + ((size_t)pair * COL_CHUNKS + chunk) * B_ROWS + rowbase;
#pragma unroll
    for (int v = 0; v < 8; ++v) pp[v] = rs[v];
  }
}

// ---------------------------------------------------------------------------
// Kernel 4: deterministic combine of the 8 column-chunk partials per row.
// ---------------------------------------------------------------------------
__global__ void reduce_kernel(const float* __restrict__ partial,
                              float* __restrict__ out) {
  const int idx = blockIdx.x * 256 + threadIdx.x;  // 0..16383
  const int pair = idx >> 13;
  const int r = idx & (B_ROWS - 1);
  float s = 0.0f;
#pragma unroll
  for (int c = 0; c < COL_CHUNKS; ++c)
    s += partial[((size_t)pair * COL_CHUNKS + c) * B_ROWS + r];
  out[idx] = s;
}

extern "C" void kernel_launch(void* const* d_in, const int* in_sizes, int n_in,
                              void* d_out, int out_size, void* d_ws, size_t ws_size,
                              hipStream_t stream) {
  const float* anchor   = (const float*)d_in[0];
  const float* positive = (const float*)d_in[1];
  const float* negative = (const float*)d_in[2];

  float* ws = (float*)d_ws;
  float* rvar    = ws;                    // 2 * 64
  float* sdiff   = ws + 128;              // 2 * NELEM
  float* diffb   = sdiff + 2 * NELEM;     // 2 * NELEM
  float* partial = diffb + 2 * NELEM;     // 2 * COL_CHUNKS * B_ROWS
  float* out = (float*)d_out;

  var_kernel<<<dim3(D_FEAT, 2), 256, 0, stream>>>(anchor, positive, negative, rvar);
  diff_kernel<<<NELEM / 256, 256, 0, stream>>>(anchor, positive, negative, rvar,
                                               sdiff, diffb);
  mdl_wmma_kernel<<<dim3(B_ROWS / 128, COL_CHUNKS, 2), 256, 0, stream>>>(
      sdiff, diffb, partial);
  reduce_kernel<<<(2 * B_ROWS) / 256, 256, 0, stream>>>(partial, out);
}